// LightGCN_48919677501960
// MI455X (gfx1250) — compile-verified
//
#include <hip/hip_runtime.h>
#include <stdint.h>

#define NUSERS   100000
#define NITEMS   50000
#define NNODES   150000
#define DIM      64
#define NEDGES   4000000               // symmetrized directed edges
#define NELEM    (NNODES * DIM)        // 9,600,000 floats
#define CHUNK    2048

typedef unsigned int  u32x4 __attribute__((ext_vector_type(4)));
typedef int           i32x8 __attribute__((ext_vector_type(8)));
typedef int           i32x4 __attribute__((ext_vector_type(4)));

// ---------------------------------------------------------------- utilities

__global__ void zero_i32_kernel(int* __restrict__ p, int n) {
    for (int i = blockIdx.x * blockDim.x + threadIdx.x; i < n;
         i += gridDim.x * blockDim.x)
        p[i] = 0;
}

__global__ void degree_kernel(const int* __restrict__ col, int* deg, int n) {
    for (int i = blockIdx.x * blockDim.x + threadIdx.x; i < n;
         i += gridDim.x * blockDim.x)
        atomicAdd(&deg[col[i]], 1);
}

// deg and dinv alias the same buffer (read int count, write float) — done in
// a separate kernel pass so there is no intra-kernel hazard.
__global__ void dinv_kernel(const int* deg, float* dinv, int n) {
    for (int i = blockIdx.x * blockDim.x + threadIdx.x; i < n;
         i += gridDim.x * blockDim.x) {
        int c = deg[i];
        dinv[i] = (c > 0) ? rsqrtf((float)c) : 0.0f;
    }
}

// x_cur = concat(user, item); out = x_cur; x_next = 0
__global__ void setup_kernel(const float4* __restrict__ uw,
                             const float4* __restrict__ iw,
                             float4* __restrict__ x_cur,
                             float4* __restrict__ x_next,
                             float4* __restrict__ out,
                             int n4, int u4) {
    for (int i = blockIdx.x * blockDim.x + threadIdx.x; i < n4;
         i += gridDim.x * blockDim.x) {
        float4 v = (i < u4) ? uw[i] : iw[i - u4];
        x_cur[i]  = v;
        out[i]    = v;
        float4 z; z.x = 0.f; z.y = 0.f; z.z = 0.f; z.w = 0.f;
        x_next[i] = z;
    }
}

// out += xd ; xs = 0  (prepare xs as the next scatter target)
__global__ void add_zero_kernel(float4* __restrict__ out,
                                const float4* __restrict__ xd,
                                float4* __restrict__ xs, int n4) {
    for (int i = blockIdx.x * blockDim.x + threadIdx.x; i < n4;
         i += gridDim.x * blockDim.x) {
        float4 o = out[i];
        float4 d = xd[i];
        o.x += d.x; o.y += d.y; o.z += d.z; o.w += d.w;
        out[i] = o;
        float4 z; z.x = 0.f; z.y = 0.f; z.z = 0.f; z.w = 0.f;
        xs[i] = z;
    }
}

// out = (out + xd) * 0.25
__global__ void finish_kernel(float4* __restrict__ out,
                              const float4* __restrict__ xd, int n4) {
    for (int i = blockIdx.x * blockDim.x + threadIdx.x; i < n4;
         i += gridDim.x * blockDim.x) {
        float4 o = out[i];
        float4 d = xd[i];
        o.x = (o.x + d.x) * 0.25f;
        o.y = (o.y + d.y) * 0.25f;
        o.z = (o.z + d.z) * 0.25f;
        o.w = (o.w + d.w) * 0.25f;
        out[i] = o;
    }
}

// ---------------------------------------------------------------- TDM issue
//
// Loads a 2D tile (R edges x 2) of the edge_index array into LDS:
//   y=0 line -> row[base .. base+R)   (edge_index + base)
//   y=1 line -> col[base .. base+R)   (edge_index + NEDGES + base, via stride)
// LDS layout after the DMA: [0,R) = rows, [R,2R) = cols.
__device__ __forceinline__ void tdm_issue_edges(const int* gbase, int base,
                                                int R, uint32_t lds_addr) {
    uint64_t ga = (uint64_t)(uintptr_t)(gbase + base);
    uint32_t t0 = (uint32_t)(NEDGES - base);   // remaining extent in dim0

    u32x4 g0;
    g0[0] = 1u;                                   // count=1, user descriptor
    g0[1] = lds_addr;                             // LDS byte address
    g0[2] = (uint32_t)ga;                         // global_addr[31:0]
    g0[3] = (uint32_t)((ga >> 32) & 0x1FFFFFFu)   // global_addr[56:32]
          | (2u << 30);                           // type = 2 ("image")

    i32x8 g1;
    g1[0] = 0x00020000;                           // data_size = 2 (4 bytes)
    g1[1] = (int)((t0 & 0xFFFFu) << 16);          // tensor_dim0[15:0]
    g1[2] = (int)((t0 >> 16) | (2u << 16));       // tensor_dim0[31:16], dim1=2
    g1[3] = (int)((uint32_t)R << 16);             // tile_dim0 = R
    g1[4] = 2;                                    // tile_dim1 = 2
    g1[5] = NEDGES;                               // tensor_dim0_stride[31:0]
    g1[6] = 0;
    g1[7] = 0;

    i32x4 z4; z4[0] = 0; z4[1] = 0; z4[2] = 0; z4[3] = 0;
    i32x8 z8; z8[0] = 0; z8[1] = 0; z8[2] = 0; z8[3] = 0;
              z8[4] = 0; z8[5] = 0; z8[6] = 0; z8[7] = 0;

    __builtin_amdgcn_tensor_load_to_lds(g0, g1, z4, z4, z8, 0);
}

// ------------------------------------------------------------- propagation
//
// Persistent blocks; each block walks chunks with stride gridDim.x,
// double-buffering edge indices in LDS via the Tensor Data Mover.
// One wave per edge: lanes 0..31 each handle 2 of the 64 embedding floats.
// Edge indices are wave-uniform -> readfirstlane so dinv loads go scalar
// (SMEM) and embedding addressing becomes SGPR-base + lane offset.
__global__ void propagate_kernel(const int* __restrict__ edges,
                                 const float* __restrict__ dinv,
                                 const float* __restrict__ xsrc,
                                 float* __restrict__ xdst) {
    __shared__ int smem[2][2 * CHUNK];

    const int wave   = threadIdx.x >> 5;
    const int lane   = threadIdx.x & 31;
    const int nwaves = blockDim.x >> 5;
    const int nChunks = (NEDGES + CHUNK - 1) / CHUNK;

    const uint32_t lds0 = (uint32_t)(uintptr_t)(&smem[0][0]);
    const uint32_t lds1 = (uint32_t)(uintptr_t)(&smem[1][0]);

    int c   = blockIdx.x;
    int buf = 0;

    if (threadIdx.x < 32 && c < nChunks) {
        int base = c * CHUNK;
        int R = min(CHUNK, NEDGES - base);
        tdm_issue_edges(edges, base, R, lds0);
    }

    for (; c < nChunks; c += gridDim.x) {
        const int base = c * CHUNK;
        const int R = min(CHUNK, NEDGES - base);

        __builtin_amdgcn_s_wait_tensorcnt(0);   // chunk c landed in LDS
        __syncthreads();

        // prefetch next chunk into the other buffer while we compute
        const int nextc = c + gridDim.x;
        if (threadIdx.x < 32 && nextc < nChunks) {
            int nbase = nextc * CHUNK;
            int nR = min(CHUNK, NEDGES - nbase);
            tdm_issue_edges(edges, nbase, nR, buf ? lds0 : lds1);
        }

        const int* rows = &smem[buf][0];
        const int* cols = &smem[buf][R];

        for (int k = wave; k < R; k += nwaves) {
            const int r  = __builtin_amdgcn_readfirstlane(rows[k]);
            const int cl = __builtin_amdgcn_readfirstlane(cols[k]);
            const float nrm = dinv[r] * dinv[cl];      // scalar loads
            const float2 v =
                *(const float2*)(xsrc + ((size_t)r << 6) + (lane << 1));
            float* dst = xdst + ((size_t)cl << 6) + (lane << 1);
            unsafeAtomicAdd(dst,     v.x * nrm);   // global_atomic_add_f32
            unsafeAtomicAdd(dst + 1, v.y * nrm);
        }
        __syncthreads();   // all waves done with buf before it is refilled
        buf ^= 1;
    }
}

// ---------------------------------------------------------------- launcher

extern "C" void kernel_launch(void* const* d_in, const int* in_sizes, int n_in,
                              void* d_out, int out_size, void* d_ws,
                              size_t ws_size, hipStream_t stream) {
    const int*   edges = (const int*)d_in[0];          // [2, 4M]: rows, cols
    const float* uw    = (const float*)d_in[1];        // [100000, 64]
    const float* iw    = (const float*)d_in[2];        // [50000, 64]
    float*       out   = (float*)d_out;                // [150000, 64]

    float* dinv = (float*)d_ws;                        // 150016 floats
    float* xA   = dinv + 150016;                       // 9.6M floats
    float* xB   = xA + NELEM;                          // 9.6M floats

    const int n4 = NELEM / 4;                          // 2,400,000
    const int u4 = NUSERS * DIM / 4;                   // 1,600,000

    // 1) degrees -> dinv (deg counts live in the dinv buffer as ints)
    zero_i32_kernel<<<256, 256, 0, stream>>>((int*)dinv, NNODES);
    degree_kernel<<<2048, 256, 0, stream>>>(edges + NEDGES, (int*)dinv, NEDGES);
    dinv_kernel<<<512, 256, 0, stream>>>((const int*)dinv, dinv, NNODES);

    // 2) xB = x0, out = x0, xA = 0
    setup_kernel<<<4096, 256, 0, stream>>>((const float4*)uw, (const float4*)iw,
                                           (float4*)xB, (float4*)xA,
                                           (float4*)out, n4, u4);

    // 3) three LightGCN layers, accumulated into out
    propagate_kernel<<<1024, 256, 0, stream>>>(edges, dinv, xB, xA);
    add_zero_kernel<<<4096, 256, 0, stream>>>((float4*)out, (const float4*)xA,
                                              (float4*)xB, n4);

    propagate_kernel<<<1024, 256, 0, stream>>>(edges, dinv, xA, xB);
    add_zero_kernel<<<4096, 256, 0, stream>>>((float4*)out, (const float4*)xB,
                                              (float4*)xA, n4);

    propagate_kernel<<<1024, 256, 0, stream>>>(edges, dinv, xB, xA);

    // 4) out = (out + xA) / 4
    finish_kernel<<<4096, 256, 0, stream>>>((float4*)out, (const float4*)xA, n4);
}